// EfficientDet_26018911879344
// MI455X (gfx1250) — compile-verified
//
#include <hip/hip_runtime.h>
#include <cmath>

#define DET_TH  0.3f
#define NMS_TH  0.4f
#define MAX_DET 100
#define NEGV    (-1e9f)
#define TILE    256

// ---------------------------------------------------------------------------
// Kernel 1: decode + clip + class-max score. Classification tile staged into
// LDS with CDNA5 async global->LDS copies (ASYNCcnt tracked).
// ---------------------------------------------------------------------------
__global__ __launch_bounds__(TILE)
void decode_score_kernel(const float* __restrict__ anchors,
                         const float* __restrict__ regression,
                         const float* __restrict__ classification,
                         float* __restrict__ boxes,    // [B*N*4] (clipped)
                         float* __restrict__ scores,   // [B*N]   (thresholded)
                         int N, int C, float W, float H)
{
    __shared__ float cls_tile[TILE * 14];

    const int b   = blockIdx.y;
    const int n0  = blockIdx.x * TILE;
    const int tid = threadIdx.x;
    const bool fullTile = (n0 + TILE <= N) && (C == 14);

    if (fullTile) {
        // Stage TILE*14 floats = 896 x 16B chunks via async copy engine.
        const int chunks = TILE * 14 / 4;
        const float* gbase = classification + ((size_t)b * N + n0) * 14;
        for (int j = tid; j < chunks; j += TILE) {
            unsigned lds = (unsigned)(size_t)(&cls_tile[j * 4]);
            unsigned long long ga = (unsigned long long)(size_t)(gbase + j * 4);
            asm volatile("global_load_async_to_lds_b128 %0, %1, off"
                         :: "v"(lds), "v"(ga) : "memory");
        }
        asm volatile("s_wait_asynccnt 0" ::: "memory");
    }
    __syncthreads();

    const int n = n0 + tid;
    if (n >= N) return;

    // ---- score = max over classes, thresholded at DET_TH ----
    float m = -INFINITY;
    if (fullTile) {
        const float* p = &cls_tile[tid * 14];
#pragma unroll
        for (int c = 0; c < 14; ++c) m = fmaxf(m, p[c]);
    } else {
        const float* p = classification + ((size_t)b * N + n) * C;
        for (int c = 0; c < C; ++c) m = fmaxf(m, p[c]);
    }
    const float s = (m > DET_TH) ? m : NEGV;

    // ---- decode (std 0.1,0.1,0.2,0.2) + clip ----
    const float4 a = ((const float4*)anchors)[n];
    const float4 r = ((const float4*)regression)[(size_t)b * N + n];
    const float wa = a.z - a.x, ha = a.w - a.y;
    const float cxa = a.x + 0.5f * wa, cya = a.y + 0.5f * ha;
    const float cx = cxa + r.x * 0.1f * wa;
    const float cy = cya + r.y * 0.1f * ha;
    const float w  = expf(r.z * 0.2f) * wa;
    const float h  = expf(r.w * 0.2f) * ha;

    const float x1 = fmaxf(cx - 0.5f * w, 0.0f);
    const float y1 = fmaxf(cy - 0.5f * h, 0.0f);
    const float x2 = fminf(cx + 0.5f * w, W);
    const float y2 = fminf(cy + 0.5f * h, H);

    ((float4*)boxes)[(size_t)b * N + n] = make_float4(x1, y1, x2, y2);
    scores[(size_t)b * N + n] = s;
}

// ---------------------------------------------------------------------------
// Block-wide argmax: wave32 shuffle reduce, then LDS tree across 32 waves.
// ---------------------------------------------------------------------------
__device__ inline void block_argmax(float best, int bidx,
                                    float* red_s, int* red_i,
                                    float* s_best, int* s_idx,
                                    int tid, int lane, int wid, int nwaves)
{
#pragma unroll
    for (int off = 16; off > 0; off >>= 1) {
        float ov = __shfl_down(best, off, 32);
        int   oi = __shfl_down(bidx, off, 32);
        if (ov > best || (ov == best && oi < bidx)) { best = ov; bidx = oi; }
    }
    if (lane == 0) { red_s[wid] = best; red_i[wid] = bidx; }
    __syncthreads();
    if (tid < 32) {
        float v = (tid < nwaves) ? red_s[tid] : -INFINITY;
        int   i = (tid < nwaves) ? red_i[tid] : 0;
#pragma unroll
        for (int off = 16; off > 0; off >>= 1) {
            float ov = __shfl_down(v, off, 32);
            int   oi = __shfl_down(i, off, 32);
            if (ov > v || (ov == v && oi < i)) { v = ov; i = oi; }
        }
        if (tid == 0) { *s_best = v; *s_idx = i; }
    }
    __syncthreads();
}

// ---------------------------------------------------------------------------
// IoU-suppress one candidate element against the chosen box; returns new score.
// ---------------------------------------------------------------------------
__device__ inline float suppress_one(float s, const float4& c, const float4& bb,
                                     float barea)
{
    if (s > 0.5f * NEGV) {  // alive (dead entries are exactly NEGV)
        const float xx1 = fmaxf(bb.x, c.x);
        const float yy1 = fmaxf(bb.y, c.y);
        const float xx2 = fminf(bb.z, c.z);
        const float yy2 = fminf(bb.w, c.w);
        const float inter = fmaxf(xx2 - xx1, 0.0f) * fmaxf(yy2 - yy1, 0.0f);
        const float area  = (c.z - c.x) * (c.w - c.y);
        const float iou   = inter / fmaxf(barea + area - inter, 1e-8f);
        if (iou > NMS_TH) s = NEGV;   // self-IoU == 1 kills the chosen box too
    }
    return s;
}

// ---------------------------------------------------------------------------
// Kernel 2: greedy NMS, one workgroup (32 waves) per batch image. Suppression
// of round k is fused with the argmax scan for round k+1 (L2-resident data).
// Score stream is processed as float4 groups (b128 loads/stores) for MLP.
// ---------------------------------------------------------------------------
__global__ __launch_bounds__(1024)
void nms_kernel(const float* __restrict__ boxes,
                float* __restrict__ scores,
                const float* __restrict__ classification,
                float* __restrict__ out,   // [B, MAX_DET, 6]
                int N, int C)
{
    const int b    = blockIdx.x;
    const int tid  = threadIdx.x;
    const int nthr = blockDim.x;
    const int lane = tid & 31;
    const int wid  = tid >> 5;
    const int nwaves = nthr >> 5;

    __shared__ float  red_s[32];
    __shared__ int    red_i[32];
    __shared__ float4 s_box;
    __shared__ float  s_best;
    __shared__ int    s_idx;

    const float4* bx = (const float4*)boxes + (size_t)b * N;
    float* sc = scores + (size_t)b * N;
    const bool vec4 = ((N & 3) == 0);
    const int  Nv   = N >> 2;

    // ---- initial argmax over thresholded scores ----
    float best = -INFINITY; int bidx = 0;
    if (vec4) {
        const float4* scv = (const float4*)sc;
        for (int g = tid; g < Nv; g += nthr) {
            const float4 s4 = scv[g];
            const int base = g << 2;
            if (s4.x > best) { best = s4.x; bidx = base;     }
            if (s4.y > best) { best = s4.y; bidx = base + 1; }
            if (s4.z > best) { best = s4.z; bidx = base + 2; }
            if (s4.w > best) { best = s4.w; bidx = base + 3; }
        }
    } else {
        for (int i = tid; i < N; i += nthr) {
            const float s = sc[i];
            if (s > best) { best = s; bidx = i; }
        }
    }
    block_argmax(best, bidx, red_s, red_i, &s_best, &s_idx, tid, lane, wid, nwaves);

    for (int it = 0; it < MAX_DET; ++it) {
        if (tid == 0) {
            const int   i = s_idx;
            const float v = s_best;
            const float4 bb = bx[i];
            s_box = bb;
            float* row = out + ((size_t)b * MAX_DET + it) * 6;
            if (v > 0.0f) {
                const float* p = classification + ((size_t)b * N + i) * C;
                float cm = p[0]; int ci = 0;
                for (int c = 1; c < C; ++c) if (p[c] > cm) { cm = p[c]; ci = c; }
                row[0] = bb.x; row[1] = bb.y; row[2] = bb.z; row[3] = bb.w;
                row[4] = (float)ci; row[5] = v;
            } else {
                for (int k = 0; k < 6; ++k) row[k] = -1.0f;
            }
        }
        __syncthreads();

        const float4 bb = s_box;
        const float barea = (bb.z - bb.x) * (bb.w - bb.y);

        // ---- fused: suppress vs chosen box AND find next argmax ----
        best = -INFINITY; bidx = 0;
        if (vec4) {
            float4* scv = (float4*)sc;
            for (int g = tid; g < Nv; g += nthr) {
                float4 s4 = scv[g];
                const int base = g << 2;
                float ss0 = s4.x, ss1 = s4.y, ss2 = s4.z, ss3 = s4.w;
                const float n0 = suppress_one(ss0, bx[base],     bb, barea);
                const float n1 = suppress_one(ss1, bx[base + 1], bb, barea);
                const float n2 = suppress_one(ss2, bx[base + 2], bb, barea);
                const float n3 = suppress_one(ss3, bx[base + 3], bb, barea);
                if (n0 != ss0 || n1 != ss1 || n2 != ss2 || n3 != ss3)
                    scv[g] = make_float4(n0, n1, n2, n3);   // exclusive owner: race-free
                if (n0 > best) { best = n0; bidx = base;     }
                if (n1 > best) { best = n1; bidx = base + 1; }
                if (n2 > best) { best = n2; bidx = base + 2; }
                if (n3 > best) { best = n3; bidx = base + 3; }
            }
        } else {
            for (int i = tid; i < N; i += nthr) {
                const float s = sc[i];
                const float ns = suppress_one(s, bx[i], bb, barea);
                if (ns != s) sc[i] = ns;
                if (ns > best) { best = ns; bidx = i; }
            }
        }
        block_argmax(best, bidx, red_s, red_i, &s_best, &s_idx, tid, lane, wid, nwaves);
    }
}

// ---------------------------------------------------------------------------
extern "C" void kernel_launch(void* const* d_in, const int* in_sizes, int n_in,
                              void* d_out, int out_size, void* d_ws, size_t ws_size,
                              hipStream_t stream)
{
    const float* anchors        = (const float*)d_in[1];
    const float* regression     = (const float*)d_in[2];
    const float* classification = (const float*)d_in[3];

    const int N = in_sizes[1] / 4;
    const int B = in_sizes[2] / (N * 4);
    const int C = in_sizes[3] / (B * N);
    const int HW = in_sizes[0] / 3;                 // inputs: [1,3,H,W], square
    const int H  = (int)(std::sqrt((double)HW) + 0.5);
    const float Wf = (float)H, Hf = (float)H;

    float* ws_boxes  = (float*)d_ws;                     // B*N*4 floats (16B aligned)
    float* ws_scores = ws_boxes + (size_t)B * N * 4;     // B*N floats

    dim3 g1((N + TILE - 1) / TILE, B);
    decode_score_kernel<<<g1, TILE, 0, stream>>>(anchors, regression, classification,
                                                 ws_boxes, ws_scores, N, C, Wf, Hf);

    nms_kernel<<<B, 1024, 0, stream>>>(ws_boxes, ws_scores, classification,
                                       (float*)d_out, N, C);
}